// MultiLayerGATv2_3547642986629
// MI455X (gfx1250) — compile-verified
//
#include <hip/hip_runtime.h>
#include <math.h>

#define N_NODES 100000
#define DIM     128
#define HEADS   4
#define CHANS   32
#define LAYERS  2
#define NEDGE   1600000
#define NEG_SLOPE 0.2f
#define LN_EPS    1e-5f

typedef __attribute__((ext_vector_type(2))) float v2f;
typedef __attribute__((ext_vector_type(8))) float v8f;

// ---------- monotone float<->uint mapping for atomic segment-max ----------
__device__ __forceinline__ unsigned f2mono(float f) {
    unsigned u = __float_as_uint(f);
    return (u & 0x80000000u) ? ~u : (u | 0x80000000u);
}
__device__ __forceinline__ float mono2f(unsigned u) {
    return __uint_as_float((u & 0x80000000u) ? (u & 0x7fffffffu) : ~u);
}

// =====================================================================
// GEMM: xl = h@Wl, xr = h@Wr.  One wave32 per 16-row strip, full 128 cols,
// both weight matrices (A tile loaded once, reused).  f32 WMMA 16x16x4.
// =====================================================================
__global__ __launch_bounds__(32) void gat_gemm_wmma(
        const float* __restrict__ h,
        const float* __restrict__ Wl, const float* __restrict__ Wr,
        float* __restrict__ xl, float* __restrict__ xr) {
    const int lane  = threadIdx.x;        // 0..31
    const int row0  = blockIdx.x * 16;
    const int rsub  = lane & 15;
    const int khalf = (lane >> 4) << 1;   // 0 or 2

    v8f accL[8] = {};
    v8f accR[8] = {};

    for (int k0 = 0; k0 < DIM; k0 += 4) {
        // A 16x4 f32: lanes 0-15 hold (M=lane, K=k0..k0+1); lanes 16-31 (M=lane-16, K=k0+2..k0+3)
        const float* ap = h + (size_t)(row0 + rsub) * DIM + (k0 + khalf);
        v2f a; a.x = ap[0]; a.y = ap[1];
        const int kr = k0 + khalf;
        #pragma unroll
        for (int t = 0; t < 8; ++t) {
            const int col = t * 16 + rsub;
            // B 4x16 f32: lanes 0-15 rows K=k0,k0+1; lanes 16-31 rows K=k0+2,k0+3
            v2f bl, br;
            bl.x = Wl[(size_t)kr * DIM + col];
            bl.y = Wl[(size_t)(kr + 1) * DIM + col];
            br.x = Wr[(size_t)kr * DIM + col];
            br.y = Wr[(size_t)(kr + 1) * DIM + col];
            accL[t] = __builtin_amdgcn_wmma_f32_16x16x4_f32(
                false, a, false, bl, (short)0, accL[t], false, false);
            accR[t] = __builtin_amdgcn_wmma_f32_16x16x4_f32(
                false, a, false, br, (short)0, accR[t], false, false);
        }
    }

    // C/D 16x16 f32: VGPR j -> row j (lanes 0-15) / row j+8 (lanes 16-31)
    const int rowAdd = (lane >> 4) << 3;
    #pragma unroll
    for (int t = 0; t < 8; ++t) {
        const int col = t * 16 + rsub;
        #pragma unroll
        for (int j = 0; j < 8; ++j) {
            const size_t idx = (size_t)(row0 + rowAdd + j) * DIM + col;
            xl[idx] = accL[t][j];
            xr[idx] = accR[t][j];
        }
    }
}

// =====================================================================
// Edge pass 1: alpha[e,h] = att[h] . leaky_relu(xl[src]+xr[dst]);
// atomic segment-max into mMono[dst,h].  One wave32 per edge, 4 ch/lane.
// =====================================================================
__global__ __launch_bounds__(256) void edge_alpha_max(
        const float* __restrict__ xl, const float* __restrict__ xr,
        const int* __restrict__ srcArr, const int* __restrict__ dstArr,
        const float* __restrict__ att,
        float* __restrict__ alpha, unsigned* __restrict__ mMono, int nEdges) {
    const int e    = (int)((blockIdx.x * (unsigned)blockDim.x + threadIdx.x) >> 5);
    const int lane = threadIdx.x & 31;
    if (e >= nEdges) return;
    int s, d;
    if (e < NEDGE) { s = srcArr[e]; d = dstArr[e]; }
    else           { s = d = e - NEDGE; }           // appended self-loops

    const float4 xlv = *(const float4*)(xl + (size_t)s * DIM + lane * 4);
    const float4 xrv = *(const float4*)(xr + (size_t)d * DIM + lane * 4);
    const float4 av  = *(const float4*)(att + lane * 4);   // att[head][c], head=lane>>3

    float e0 = xlv.x + xrv.x; e0 = (e0 > 0.f) ? e0 : NEG_SLOPE * e0;
    float e1 = xlv.y + xrv.y; e1 = (e1 > 0.f) ? e1 : NEG_SLOPE * e1;
    float e2 = xlv.z + xrv.z; e2 = (e2 > 0.f) ? e2 : NEG_SLOPE * e2;
    float e3 = xlv.w + xrv.w; e3 = (e3 > 0.f) ? e3 : NEG_SLOPE * e3;
    float p = e0 * av.x + e1 * av.y + e2 * av.z + e3 * av.w;

    // reduce across 8 lanes of this head
    p += __shfl_xor(p, 1);
    p += __shfl_xor(p, 2);
    p += __shfl_xor(p, 4);

    if ((lane & 7) == 0) {
        const int head = lane >> 3;
        alpha[(size_t)e * HEADS + head] = p;
        atomicMax(&mMono[(size_t)d * HEADS + head], f2mono(p));
    }
}

// =====================================================================
// Edge pass 2: ea = exp(alpha - m[dst]); s[dst,h] += ea;
// acc[dst,:] += ea * xl[src,:]  (unnormalized; divided in finalize).
// =====================================================================
__global__ __launch_bounds__(256) void edge_exp_scatter(
        const float* __restrict__ xl,
        const int* __restrict__ srcArr, const int* __restrict__ dstArr,
        const float* __restrict__ alpha, const unsigned* __restrict__ mMono,
        float* __restrict__ sSum, float* __restrict__ acc, int nEdges) {
    const int e    = (int)((blockIdx.x * (unsigned)blockDim.x + threadIdx.x) >> 5);
    const int lane = threadIdx.x & 31;
    if (e >= nEdges) return;
    int s, d;
    if (e < NEDGE) { s = srcArr[e]; d = dstArr[e]; }
    else           { s = d = e - NEDGE; }
    const int head = lane >> 3;

    const float m  = mono2f(mMono[(size_t)d * HEADS + head]);
    const float al = alpha[(size_t)e * HEADS + head];
    const float ea = __expf(al - m);

    if ((lane & 7) == 0)
        atomicAdd(&sSum[(size_t)d * HEADS + head], ea);

    const float4 xlv = *(const float4*)(xl + (size_t)s * DIM + lane * 4);
    float* ap = acc + (size_t)d * DIM + lane * 4;
    atomicAdd(ap + 0, ea * xlv.x);
    atomicAdd(ap + 1, ea * xlv.y);
    atomicAdd(ap + 2, ea * xlv.z);
    atomicAdd(ap + 3, ea * xlv.w);
}

// =====================================================================
// Node finalize: g = acc/s + bias; LN(g)*gamma+beta; hnext = hprev + elu.
// One wave32 per node; warp shuffles for mean/var.
// =====================================================================
__global__ __launch_bounds__(256) void node_finalize(
        const float* __restrict__ acc, const float* __restrict__ sSum,
        const float* __restrict__ bias, const float* __restrict__ gamma,
        const float* __restrict__ beta,
        const float* __restrict__ hprev, float* __restrict__ hnext) {
    const int node = (int)((blockIdx.x * (unsigned)blockDim.x + threadIdx.x) >> 5);
    const int lane = threadIdx.x & 31;
    if (node >= N_NODES) return;
    const int head = lane >> 3;

    const float inv = 1.0f / sSum[(size_t)node * HEADS + head];
    const float4 av = *(const float4*)(acc  + (size_t)node * DIM + lane * 4);
    const float4 bv = *(const float4*)(bias + lane * 4);
    float g0 = av.x * inv + bv.x;
    float g1 = av.y * inv + bv.y;
    float g2 = av.z * inv + bv.z;
    float g3 = av.w * inv + bv.w;

    float sum = g0 + g1 + g2 + g3;
    float sq  = g0 * g0 + g1 * g1 + g2 * g2 + g3 * g3;
    #pragma unroll
    for (int m = 1; m < 32; m <<= 1) {
        sum += __shfl_xor(sum, m);
        sq  += __shfl_xor(sq,  m);
    }
    const float mu   = sum * (1.0f / DIM);
    const float var  = sq * (1.0f / DIM) - mu * mu;
    const float rstd = rsqrtf(var + LN_EPS);

    const float4 gv = *(const float4*)(gamma + lane * 4);
    const float4 be = *(const float4*)(beta  + lane * 4);
    const float4 hp = *(const float4*)(hprev + (size_t)node * DIM + lane * 4);

    float y0 = (g0 - mu) * rstd * gv.x + be.x;
    float y1 = (g1 - mu) * rstd * gv.y + be.y;
    float y2 = (g2 - mu) * rstd * gv.z + be.z;
    float y3 = (g3 - mu) * rstd * gv.w + be.w;
    y0 = (y0 > 0.f) ? y0 : (__expf(y0) - 1.0f);
    y1 = (y1 > 0.f) ? y1 : (__expf(y1) - 1.0f);
    y2 = (y2 > 0.f) ? y2 : (__expf(y2) - 1.0f);
    y3 = (y3 > 0.f) ? y3 : (__expf(y3) - 1.0f);

    float4 out;
    out.x = hp.x + y0; out.y = hp.y + y1; out.z = hp.z + y2; out.w = hp.w + y3;
    *(float4*)(hnext + (size_t)node * DIM + lane * 4) = out;
}

// =====================================================================
extern "C" void kernel_launch(void* const* d_in, const int* in_sizes, int n_in,
                              void* d_out, int out_size, void* d_ws, size_t ws_size,
                              hipStream_t stream) {
    const float* x     = (const float*)d_in[0];
    const int*   ei    = (const int*)  d_in[1];   // (2, E)
    const float* Wl    = (const float*)d_in[2];   // (L, D, D)
    const float* Wr    = (const float*)d_in[3];
    const float* att   = (const float*)d_in[4];   // (L, H, C)
    const float* bias  = (const float*)d_in[5];   // (L, D)
    const float* gamma = (const float*)d_in[6];
    const float* beta  = (const float*)d_in[7];
    float* out = (float*)d_out;

    const size_t ND = (size_t)N_NODES * DIM;
    const int    E2 = NEDGE + N_NODES;

    float*    xl     = (float*)d_ws;
    float*    xr     = xl + ND;
    float*    hbuf   = xr + ND;                     // layer-1 acc, then h1
    float*    alphaB = hbuf + ND;                   // E2 * HEADS
    unsigned* mB     = (unsigned*)(alphaB + (size_t)E2 * HEADS);
    float*    sB     = (float*)(mB + (size_t)N_NODES * HEADS);

    const int* srcA = ei;
    const int* dstA = ei + NEDGE;

    const int gemmGrid = N_NODES / 16;              // 6250
    const int edgeGrid = (E2 + 7) / 8;              // 8 waves / 256-thread block
    const int nodeGrid = (N_NODES + 7) / 8;

    for (int l = 0; l < LAYERS; ++l) {
        const float* hin  = (l == 0) ? x    : hbuf;
        float*       accp = (l == 0) ? hbuf : out;  // layer-2 accumulates in d_out
        float*       hnxt = (l == 0) ? hbuf : out;

        hipMemsetAsync(accp, 0, ND * sizeof(float), stream);
        hipMemsetAsync(mB,   0, (size_t)N_NODES * HEADS * sizeof(unsigned), stream);
        hipMemsetAsync(sB,   0, (size_t)N_NODES * HEADS * sizeof(float), stream);

        gat_gemm_wmma<<<gemmGrid, 32, 0, stream>>>(
            hin, Wl + (size_t)l * DIM * DIM, Wr + (size_t)l * DIM * DIM, xl, xr);

        edge_alpha_max<<<edgeGrid, 256, 0, stream>>>(
            xl, xr, srcA, dstA, att + (size_t)l * HEADS * CHANS, alphaB, mB, E2);

        edge_exp_scatter<<<edgeGrid, 256, 0, stream>>>(
            xl, srcA, dstA, alphaB, mB, sB, accp, E2);

        node_finalize<<<nodeGrid, 256, 0, stream>>>(
            accp, sB, bias + (size_t)l * DIM, gamma + (size_t)l * DIM,
            beta + (size_t)l * DIM, hin, hnxt);
    }
}